// Foam_74826920231100
// MI455X (gfx1250) — compile-verified
//
#include <hip/hip_runtime.h>
#include <math.h>

// Sizes from the reference
#define BATCH 16
#define NB    128           // N bubbles
#define DD    128           // D dim
#define ND    (NB*DD)       // per-batch state stride (N*D = 16384)
#define NN    (NB*NB)       // 16384
#define EPSF  1e-10f

typedef float v2f __attribute__((ext_vector_type(2)));
typedef float v8f __attribute__((ext_vector_type(8)));

// ---------------------------------------------------------------------------
// f32 WMMA helpers (V_WMMA_F32_16X16X4_F32, wave32, one wave per 16x16 tile)
// A-layout (16x4 f32): lane l -> m = l&15, k-pair base = 2*(l>>4); vgpr0/1 = k0/k1
// B-layout (4x16 f32): mirrored; C/D (16x16): vgpr v -> row v + 8*(l>>4), col l&15
// ---------------------------------------------------------------------------
__device__ __forceinline__ v8f wmma_zero() {
  v8f a;
#pragma unroll
  for (int i = 0; i < 8; ++i) a[i] = 0.0f;
  return a;
}

// C(16x16) += A(16xK) * B(Kx16), generic element strides so we can feed
// transposed operands without copies.
__device__ __forceinline__ v8f wmma_tile_f32(const float* __restrict__ A, int rsA, int csA,
                                             const float* __restrict__ Bm, int rsB, int csB,
                                             int K, v8f acc) {
  const int lane = threadIdx.x & 31;
  const int half = lane >> 4;     // 0 or 1
  const int mn   = lane & 15;
  for (int k = 0; k < K; k += 4) {
    const int k0 = k + 2 * half;
    v2f a, b;
    a.x = A[mn * rsA + (k0 + 0) * csA];
    a.y = A[mn * rsA + (k0 + 1) * csA];
    b.x = Bm[(k0 + 0) * rsB + mn * csB];
    b.y = Bm[(k0 + 1) * rsB + mn * csB];
    acc = __builtin_amdgcn_wmma_f32_16x16x4_f32(false, a, false, b, (short)0, acc,
                                                false, false);
  }
  return acc;
}

// ---------------------------------------------------------------------------
// 1) Cayley transform: bases[n] = (I+A)^-1 (I-A), A = L - L^T.
//    Gauss-Jordan on a 128x256 augmented matrix in global scratch,
//    one workgroup (256 threads) per bubble. fac[p]=0 makes row p a no-op.
// ---------------------------------------------------------------------------
__global__ void k_cayley(const float* __restrict__ L, float* __restrict__ aug,
                         float* __restrict__ bases) {
  const int n   = blockIdx.x;
  const int tid = threadIdx.x;                       // 0..255
  float* Ag = aug + (size_t)n * (DD * 2 * DD);       // 128 x 256
  const float* Ln = L + (size_t)n * DD * DD;
  __shared__ float prow[2 * DD];
  __shared__ float fac[DD];

  for (int idx = tid; idx < DD * 2 * DD; idx += 256) {
    int i = idx >> 8;
    int j = idx & 255;
    float v;
    if (j < DD) {
      float a = Ln[i * DD + j] - Ln[j * DD + i];
      v = (i == j ? 1.0f : 0.0f) + a;                // I + A
    } else {
      int jj = j - DD;
      float a = Ln[i * DD + jj] - Ln[jj * DD + i];
      v = (i == jj ? 1.0f : 0.0f) - a;               // I - A
    }
    Ag[idx] = v;
  }
  __syncthreads();

  for (int p = 0; p < DD; ++p) {
    prow[tid] = Ag[p * 256 + tid];
    __syncthreads();
    float inv = 1.0f / prow[p];
    __syncthreads();
    float v = prow[tid] * inv;                       // scale pivot row
    prow[tid] = v;
    Ag[p * 256 + tid] = v;
    if (tid < DD) fac[tid] = (tid == p) ? 0.0f : Ag[tid * 256 + p];
    __syncthreads();
    for (int idx = tid; idx < DD * 256; idx += 256) {
      int i = idx >> 8;
      Ag[idx] -= fac[i] * prow[idx & 255];
    }
    __syncthreads();
  }
  for (int idx = tid; idx < DD * DD; idx += 256) {
    int i = idx >> 7, j = idx & 127;
    bases[(size_t)n * DD * DD + idx] = Ag[i * 256 + DD + j];
  }
}

// ---------------------------------------------------------------------------
// 2) measurements[b,n,k] = sum_d x[b,d] * bases[n,d,k]  (M=16 == BATCH)
//    grid(ktile=8, n=128), one wave per 16x16 output tile.
// ---------------------------------------------------------------------------
__global__ void k_measure(const float* __restrict__ x, const float* __restrict__ bases,
                          float* __restrict__ meas) {
  const int kt = blockIdx.x;
  const int n  = blockIdx.y;
  v8f acc = wmma_zero();
  const float* Bm = bases + (size_t)n * DD * DD + kt * 16;   // B[d][col], rs=DD
  acc = wmma_tile_f32(x, DD, 1, Bm, DD, 1, DD, acc);
  float* C = meas + (size_t)n * DD + kt * 16;
  const int lane = threadIdx.x & 31;
  const int half = lane >> 4, col = lane & 15;
#pragma unroll
  for (int v = 0; v < 8; ++v)
    C[(size_t)(v + 8 * half) * ND + col] = acc[v];           // batch stride = N*D
}

// ---------------------------------------------------------------------------
// 3) Gram of flattened bases: G = Bf * Bf^T, Bf = [128, 16384]
// ---------------------------------------------------------------------------
__global__ void k_gram(const float* __restrict__ bases, float* __restrict__ G) {
  const int jt = blockIdx.x, it = blockIdx.y;
  v8f acc = wmma_zero();
  const float* A  = bases + (size_t)it * 16 * DD * DD;        // rows i, rs=16384
  const float* Bm = bases + (size_t)jt * 16 * DD * DD;        // B[k][j]=Bf[j][k]
  acc = wmma_tile_f32(A, DD * DD, 1, Bm, 1, DD * DD, DD * DD, acc);
  float* C = G + it * 16 * NB + jt * 16;
  const int lane = threadIdx.x & 31;
  const int half = lane >> 4, col = lane & 15;
#pragma unroll
  for (int v = 0; v < 8; ++v) C[(v + 8 * half) * NB + col] = acc[v];
}

// ---------------------------------------------------------------------------
// 4) tension[i,j] = sqrt(Gii+Gjj-2Gij + 1e-8); w = mask * softmax(-t/temp)
// ---------------------------------------------------------------------------
__global__ void k_tension(const float* __restrict__ G, const float* __restrict__ tempP,
                          float* __restrict__ tension, float* __restrict__ wmat) {
  const int i = blockIdx.x, j = threadIdx.x;
  __shared__ float red[NB];
  float Gii = G[i * NB + i];
  float Gjj = G[j * NB + j];
  float t = sqrtf(fmaxf(Gii + Gjj - 2.0f * G[i * NB + j], 0.0f) + 1e-8f);
  tension[i * NB + j] = t;
  float temp = fmaxf(fabsf(tempP[0]), 0.01f);
  float sc = -t / temp;
  red[j] = sc; __syncthreads();
  for (int s = 64; s > 0; s >>= 1) { if (j < s) red[j] = fmaxf(red[j], red[j + s]); __syncthreads(); }
  float mx = red[0]; __syncthreads();
  float e = expf(sc - mx);
  red[j] = e; __syncthreads();
  for (int s = 64; s > 0; s >>= 1) { if (j < s) red[j] += red[j + s]; __syncthreads(); }
  float sum = red[0];
  wmat[i * NB + j] = (j == i) ? 0.0f : (e / sum);
}

// ---------------------------------------------------------------------------
// 5a) per-(b,n) squared norms of state rows
// ---------------------------------------------------------------------------
__global__ void k_norms(const float* __restrict__ st, float* __restrict__ nn2) {
  int id = blockIdx.x * blockDim.x + threadIdx.x;   // 0..2047
  const float* s = st + (size_t)id * DD;
  float acc = 0.f;
  for (int d = 0; d < DD; ++d) { float v = s[d]; acc += v * v; }
  nn2[id] = acc;
}

// ---------------------------------------------------------------------------
// 5b) g[b,n,m] = (cos - tgt) * w[n,m] / (||s_n - s_m|| + eps); fused epilogue
//     on the batched S*S^T WMMA GEMM (||diff||^2 = nn+mm-2dot).
// ---------------------------------------------------------------------------
__global__ void k_gmat(const float* __restrict__ st, const float* __restrict__ nn2,
                       const float* __restrict__ wmat, const float* __restrict__ tgtP,
                       float* __restrict__ g) {
  const int mt = blockIdx.x, nt = blockIdx.y, b = blockIdx.z;
  v8f acc = wmma_zero();
  const float* A  = st + (size_t)b * ND + nt * 16 * DD;       // S rows n
  const float* Bm = st + (size_t)b * ND + mt * 16 * DD;       // B[d][m] = S[m][d]
  acc = wmma_tile_f32(A, DD, 1, Bm, 1, DD, DD, acc);
  const float tgt = tgtP[0];
  const int lane = threadIdx.x & 31;
  const int half = lane >> 4, col = lane & 15;
  const int m = mt * 16 + col;
  float bn2 = nn2[b * NB + m];
#pragma unroll
  for (int v = 0; v < 8; ++v) {
    int n = nt * 16 + v + 8 * half;
    float an2 = nn2[b * NB + n];
    float dot = acc[v];
    float cosv = dot / ((sqrtf(an2) + EPSF) * (sqrtf(bn2) + EPSF));
    float d2 = fmaxf(an2 + bn2 - 2.0f * dot, 0.0f);
    float rinv = 1.0f / (sqrtf(d2) + EPSF);
    g[(size_t)b * NN + n * NB + m] = (cosv - tgt) * wmat[n * NB + m] * rinv;
  }
}

__global__ void k_rowsum(const float* __restrict__ g, float* __restrict__ rsum) {
  int id = blockIdx.x * blockDim.x + threadIdx.x;   // (b,n) flat
  const float* row = g + (size_t)id * NB;
  float s = 0.f;
  for (int m = 0; m < NB; ++m) s += row[m];
  rsum[id] = s;
}

// ---------------------------------------------------------------------------
// 5c) state' = state + step*(state*rowsum - g@state); WMMA GEMM + fused axpy
// ---------------------------------------------------------------------------
__global__ void k_update(const float* __restrict__ g, const float* __restrict__ st,
                         const float* __restrict__ rsum, const float* __restrict__ stepP,
                         float* __restrict__ stnew) {
  const int dt = blockIdx.x, nt = blockIdx.y, b = blockIdx.z;
  v8f acc = wmma_zero();
  const float* A  = g  + (size_t)b * NN + nt * 16 * NB;
  const float* Bm = st + (size_t)b * ND + dt * 16;            // B[m][d], rs=DD
  acc = wmma_tile_f32(A, NB, 1, Bm, DD, 1, NB, acc);
  float step = fminf(fmaxf(fabsf(stepP[0]), 0.001f), 0.5f);
  const int lane = threadIdx.x & 31;
  const int half = lane >> 4, col = lane & 15;
  const int d = dt * 16 + col;
#pragma unroll
  for (int v = 0; v < 8; ++v) {
    int n = nt * 16 + v + 8 * half;
    size_t off = (size_t)b * ND + n * DD + d;
    float s = st[off];
    stnew[off] = s + step * (s * rsum[b * NB + n] - acc[v]);
  }
}

// ---------------------------------------------------------------------------
// 6) expressions[b,n,k] = sum_d eq[b,n,d]*bases[n,k,d]
// ---------------------------------------------------------------------------
__global__ void k_expr(const float* __restrict__ eq, const float* __restrict__ bases,
                       float* __restrict__ expr) {
  const int kt = blockIdx.x, n = blockIdx.y;
  v8f acc = wmma_zero();
  const float* A  = eq + (size_t)n * DD;                      // rows b, rs=N*D
  const float* Bm = bases + (size_t)n * DD * DD + kt * 16 * DD; // B[d][col]=bases[n][k][d]
  acc = wmma_tile_f32(A, ND, 1, Bm, 1, DD, DD, acc);
  float* C = expr + (size_t)n * DD + kt * 16;
  const int lane = threadIdx.x & 31;
  const int half = lane >> 4, col = lane & 15;
#pragma unroll
  for (int v = 0; v < 8; ++v)
    C[(size_t)(v + 8 * half) * ND + col] = acc[v];
}

__global__ void k_avg(const float* __restrict__ expr, float* __restrict__ avg) {
  int id = blockIdx.x * blockDim.x + threadIdx.x;  // b*128+k
  int b = id >> 7, k = id & 127;
  float s = 0.f;
  for (int n = 0; n < NB; ++n) s += expr[(size_t)b * ND + n * DD + k];
  avg[id] = s * (1.0f / NB);
}

// ---------------------------------------------------------------------------
// 7) m_norm rows; rho[b] = m_norm^T m_norm / N (WMMA)
// ---------------------------------------------------------------------------
__global__ void k_mnorm(const float* __restrict__ st, float* __restrict__ mn) {
  const int id = blockIdx.x;     // (b,n) flat
  const int t  = threadIdx.x;    // 0..127
  __shared__ float red[DD];
  float v = st[(size_t)id * DD + t];
  red[t] = v * v; __syncthreads();
  for (int s = 64; s > 0; s >>= 1) { if (t < s) red[t] += red[t + s]; __syncthreads(); }
  float inv = 1.0f / (sqrtf(red[0]) + EPSF);
  mn[(size_t)id * DD + t] = v * inv;
}

__global__ void k_rho(const float* __restrict__ mn, float* __restrict__ rho) {
  const int et = blockIdx.x, dt = blockIdx.y, b = blockIdx.z;
  v8f acc = wmma_zero();
  const float* A  = mn + (size_t)b * ND + dt * 16;   // A[d][n]=mn[n][d]: rs=1, cs=DD
  const float* Bm = mn + (size_t)b * ND + et * 16;   // B[n][e]: rs=DD, cs=1
  acc = wmma_tile_f32(A, 1, DD, Bm, DD, 1, NB, acc);
  const int lane = threadIdx.x & 31;
  const int half = lane >> 4, col = lane & 15;
#pragma unroll
  for (int v = 0; v < 8; ++v)
    rho[(size_t)b * DD * DD + (dt * 16 + v + 8 * half) * DD + et * 16 + col] =
        acc[v] * (1.0f / NB);
}

// ---------------------------------------------------------------------------
// 8) eigenvalues of symmetric rho[b] via parallel two-sided Jacobi.
//    64 threads, each owns one disjoint rotation pair per round (angle stays
//    in registers), matrix in 64KB LDS, then odd-even sort (ascending).
// ---------------------------------------------------------------------------
__global__ void k_eig(const float* __restrict__ rho, float* __restrict__ dist) {
  __shared__ float Am[NB * NB];                      // exactly 64 KB
  const int b = blockIdx.x;
  const int tid = threadIdx.x;                       // 0..63
  const float* R = rho + (size_t)b * NB * NB;
  for (int i = tid; i < NB * NB; i += 64) Am[i] = R[i];
  __syncthreads();

  for (int sweep = 0; sweep < 15; ++sweep) {
    for (int r = 0; r < 127; ++r) {
      // round-robin disjoint pairing: fix 127, rotate 0..126
      int p, q;
      if (tid == 0) { p = 127; q = r % 127; }
      else { p = (r + tid) % 127; q = (r + 127 - tid) % 127; }
      float app = Am[p * NB + p];
      float aqq = Am[q * NB + q];
      float apq = Am[p * NB + q];
      float c, s;
      if (fabsf(apq) < 1e-30f) { c = 1.0f; s = 0.0f; }
      else {
        float tau = (aqq - app) / (2.0f * apq);
        float tt  = copysignf(1.0f, tau) / (fabsf(tau) + sqrtf(1.0f + tau * tau));
        c = rsqrtf(1.0f + tt * tt);
        s = tt * c;
      }
      __syncthreads();
      for (int j = 0; j < NB; ++j) {                 // rows p,q (disjoint per thread)
        float ap = Am[p * NB + j], aq = Am[q * NB + j];
        Am[p * NB + j] = c * ap - s * aq;
        Am[q * NB + j] = s * ap + c * aq;
      }
      __syncthreads();
      for (int i = 0; i < NB; ++i) {                 // cols p,q (disjoint per thread)
        float ap = Am[i * NB + p], aq = Am[i * NB + q];
        Am[i * NB + p] = c * ap - s * aq;
        Am[i * NB + q] = s * ap + c * aq;
      }
      __syncthreads();
    }
  }
  // diag -> Am[0..127] (clamped), ascending odd-even sort, normalize
  float d0 = fmaxf(Am[tid * NB + tid], 1e-12f);
  float d1 = fmaxf(Am[(tid + 64) * NB + (tid + 64)], 1e-12f);
  __syncthreads();
  Am[tid] = d0; Am[tid + 64] = d1;
  __syncthreads();
  for (int phase = 0; phase < NB; ++phase) {
    int i = 2 * tid + (phase & 1);
    if (i + 1 < NB) {
      float a = Am[i], bb = Am[i + 1];
      if (a > bb) { Am[i] = bb; Am[i + 1] = a; }
    }
    __syncthreads();
  }
  if (tid == 0) {
    float s = 0.f;
    for (int i = 0; i < NB; ++i) s += Am[i];
    Am[NB] = s;
  }
  __syncthreads();
  float inv = 1.0f / Am[NB];
  dist[b * NB + tid]      = Am[tid] * inv;
  dist[b * NB + tid + 64] = Am[tid + 64] * inv;
}

// ---------------------------------------------------------------------------
// Host-side orchestration
// ---------------------------------------------------------------------------
extern "C" void kernel_launch(void* const* d_in, const int* in_sizes, int n_in,
                              void* d_out, int out_size, void* d_ws, size_t ws_size,
                              hipStream_t stream) {
  const float* x    = (const float*)d_in[0];
  const float* L    = (const float*)d_in[1];
  const float* temp = (const float*)d_in[2];
  const float* tgt  = (const float*)d_in[3];
  const float* stp  = (const float*)d_in[4];
  float* out = (float*)d_out;

  // Output layout (floats), reference return order
  const size_t OFF_AVG  = 0;                       // [16,128]
  const size_t OFF_DIST = OFF_AVG  + 2048;         // [16,128]
  const size_t OFF_RHO  = OFF_DIST + 2048;         // [16,128,128]
  const size_t OFF_MEAS = OFF_RHO  + 262144;       // [16,128,128]
  const size_t OFF_EQ   = OFF_MEAS + 262144;       // [16,128,128]
  const size_t OFF_EXPR = OFF_EQ   + 262144;       // [16,128,128]
  const size_t OFF_TEN  = OFF_EXPR + 262144;       // [128,128]

  // Workspace layout: bases persists; solver scratch region is reused for
  // all post-solve buffers (solver finishes before they are written).
  // Total requirement: (2,097,152 + 4,194,304) floats = ~25.2 MB.
  float* ws     = (float*)d_ws;
  float* bases  = ws;                               // 128*128*128
  float* region = ws + 2097152;
  float* aug    = region;                           // 128 * (128*256)
  float* stateA = region;
  float* stateB = stateA + (size_t)BATCH * ND;
  float* G      = stateB + (size_t)BATCH * ND;
  float* wmat   = G + NN;
  float* nn2    = wmat + NN;
  float* gmat   = nn2 + BATCH * NB;
  float* rsum   = gmat + (size_t)BATCH * NN;
  float* mnorm  = rsum + BATCH * NB;

  k_cayley<<<dim3(NB), dim3(256), 0, stream>>>(L, aug, bases);
  k_measure<<<dim3(8, NB), dim3(32), 0, stream>>>(x, bases, out + OFF_MEAS);
  k_gram<<<dim3(8, 8), dim3(32), 0, stream>>>(bases, G);
  k_tension<<<dim3(NB), dim3(NB), 0, stream>>>(G, temp, out + OFF_TEN, wmat);

  hipMemcpyAsync(stateA, out + OFF_MEAS, sizeof(float) * (size_t)BATCH * ND,
                 hipMemcpyDeviceToDevice, stream);

  float* cur = stateA; float* nxt = stateB;
  for (int it = 0; it < 5; ++it) {
    k_norms<<<dim3(8), dim3(256), 0, stream>>>(cur, nn2);
    k_gmat<<<dim3(8, 8, BATCH), dim3(32), 0, stream>>>(cur, nn2, wmat, tgt, gmat);
    k_rowsum<<<dim3(8), dim3(256), 0, stream>>>(gmat, rsum);
    k_update<<<dim3(8, 8, BATCH), dim3(32), 0, stream>>>(gmat, cur, rsum, stp, nxt);
    float* t = cur; cur = nxt; nxt = t;
  }

  hipMemcpyAsync(out + OFF_EQ, cur, sizeof(float) * (size_t)BATCH * ND,
                 hipMemcpyDeviceToDevice, stream);
  k_expr<<<dim3(8, NB), dim3(32), 0, stream>>>(cur, bases, out + OFF_EXPR);
  k_avg<<<dim3(8), dim3(256), 0, stream>>>(out + OFF_EXPR, out + OFF_AVG);
  k_mnorm<<<dim3(BATCH * NB), dim3(DD), 0, stream>>>(cur, mnorm);
  k_rho<<<dim3(8, 8, BATCH), dim3(32), 0, stream>>>(mnorm, out + OFF_RHO);
  k_eig<<<dim3(BATCH), dim3(64), 0, stream>>>(out + OFF_RHO, out + OFF_DIST);

  (void)in_sizes; (void)n_in; (void)out_size; (void)ws_size;
}